// GAT_44839458571021
// MI455X (gfx1250) — compile-verified
//
#include <hip/hip_runtime.h>
#include <math.h>

#define NNODES 50000
#define NEDGES 800000
#define NGRAPH 256
#define FIN    128
#define H1C    12
#define D1C    32
#define F1     (H1C * D1C)   // 384
#define H2C    1
#define D2C    64
#define F2     (H2C * D2C)   // 64
#define ETOT   (NEDGES + NNODES)  // edges + self loops

typedef float v2f __attribute__((ext_vector_type(2)));
typedef float v8f __attribute__((ext_vector_type(8)));

// ---- monotone float<->uint key for atomic max on floats ----
__device__ __forceinline__ unsigned fkey(float f) {
  unsigned u = __float_as_uint(f);
  return (u & 0x80000000u) ? ~u : (u | 0x80000000u);
}
__device__ __forceinline__ float fdecode(unsigned u) {
  unsigned b = (u & 0x80000000u) ? (u & 0x7fffffffu) : ~u;
  return __uint_as_float(b);
}

// ============================================================
// WMMA fp32 GEMM: C[M,Nc] = A[M,K] @ B[K,Nc], row-major.
// One 16x16 C tile per wave via V_WMMA_F32_16X16X4_F32.
// Block = 128 threads = 4 waves -> 4 consecutive N-tiles.
// Grid  = (M/16, Nc/64). M%16==0, K%4==0, Nc%64==0 guaranteed.
// A frag (16x4): lane l<16 holds row m0+l, K = k0+{0,1};
//                lane l>=16 holds row m0+(l-16), K = k0+{2,3}.
// B frag (4x16): VGPR0 = row k0 (+2 for hi half) striped over lanes.
// C (16x16): vgpr r -> row m0 + r + 8*(lane>=16), col n0 + (lane&15).
// ============================================================
__global__ void __launch_bounds__(128)
wmma_gemm_f32(const float* __restrict__ A, const float* __restrict__ B,
              float* __restrict__ C, int K, int Ncols) {
  const int lane = threadIdx.x & 31;
  const int wave = threadIdx.x >> 5;
  const int l    = lane & 15;
  const int half = lane >> 4;          // 0 or 1
  const int m0   = blockIdx.x << 4;
  const int n0   = ((blockIdx.y << 2) + wave) << 4;

  v8f c = {};
  const float* arow = A + (size_t)(m0 + l) * K + (half << 1);
  const float* bcol = B + n0 + l;

  for (int k0 = 0; k0 < K; k0 += 4) {
    v2f a = *(const v2f*)(arow + k0);               // 8B aligned (K even)
    const int kb = k0 + (half << 1);
    v2f b;
    b.x = bcol[(size_t)kb * Ncols];
    b.y = bcol[(size_t)(kb + 1) * Ncols];
    c = __builtin_amdgcn_wmma_f32_16x16x4_f32(
        /*neg_a=*/false, a, /*neg_b=*/false, b,
        /*c_mod=*/(short)0, c, /*reuse_a=*/false, /*reuse_b=*/false);
  }

  float* crow = C + (size_t)(m0 + (half << 3)) * Ncols + n0 + l;
#pragma unroll
  for (int r = 0; r < 8; ++r) crow[(size_t)r * Ncols] = c[r];
}

// a_src[n,h] = sum_d h[n,h,d]*att_src[h,d]; same for dst. 1 thread/(n,h).
__global__ void att_coeff(const float* __restrict__ h,
                          const float* __restrict__ att_src,
                          const float* __restrict__ att_dst,
                          float* __restrict__ asrc, float* __restrict__ adst,
                          int H, int D) {
  int i = blockIdx.x * blockDim.x + threadIdx.x;
  if (i >= NNODES * H) return;
  int n = i / H, hh = i - n * H;
  const float* hp = h + (size_t)n * H * D + hh * D;
  const float* ws = att_src + hh * D;
  const float* wd = att_dst + hh * D;
  float s = 0.f, d = 0.f;
  for (int j = 0; j < D; ++j) { float v = hp[j]; s += v * ws[j]; d += v * wd[j]; }
  asrc[i] = s; adst[i] = d;
}

// pass 1: per (edge,head): e = leakyrelu(asrc[src]+adst[dst]); atomic max into mkey[dst]
__global__ void edge_logit_max(const int* __restrict__ srcI, const int* __restrict__ dstI,
                               const float* __restrict__ asrc, const float* __restrict__ adst,
                               float* __restrict__ e, unsigned* __restrict__ mkey, int H) {
  int i = blockIdx.x * blockDim.x + threadIdx.x;
  if (i >= ETOT * H) return;
  int eid = i / H, hh = i - eid * H;
  int s, d;
  if (eid < NEDGES) { s = srcI[eid]; d = dstI[eid]; } else { s = eid - NEDGES; d = s; }
  float v = asrc[s * H + hh] + adst[d * H + hh];
  v = (v > 0.f) ? v : 0.2f * v;
  e[i] = v;
  atomicMax(&mkey[d * H + hh], fkey(v));
}

// pass 2: p = exp(e - m[dst]); e <- p; atomic add into ssum[dst]
__global__ void edge_exp_sum(const int* __restrict__ dstI, float* __restrict__ e,
                             const unsigned* __restrict__ mkey, float* __restrict__ ssum,
                             int H) {
  int i = blockIdx.x * blockDim.x + threadIdx.x;
  if (i >= ETOT * H) return;
  int eid = i / H, hh = i - eid * H;
  int d = (eid < NEDGES) ? dstI[eid] : (eid - NEDGES);
  float p = __expf(e[i] - fdecode(mkey[d * H + hh]));
  e[i] = p;
  atomicAdd(&ssum[d * H + hh], p);
}

// pass 3: alpha = p / s[dst]   (folded so the big agg pass skips the divide)
__global__ void edge_alpha(const int* __restrict__ dstI, float* __restrict__ e,
                           const float* __restrict__ ssum, int H) {
  int i = blockIdx.x * blockDim.x + threadIdx.x;
  if (i >= ETOT * H) return;
  int eid = i / H, hh = i - eid * H;
  int d = (eid < NEDGES) ? dstI[eid] : (eid - NEDGES);
  e[i] = e[i] / ssum[d * H + hh];
}

// pass 4: out[dst, f] += alpha[e, f/D] * h[src, f]; 1 thread/(edge,feature)
__global__ void edge_agg(const int* __restrict__ srcI, const int* __restrict__ dstI,
                         const float* __restrict__ alpha, const float* __restrict__ h,
                         float* __restrict__ out, int H, int D) {
  const int F = H * D;
  int i = blockIdx.x * blockDim.x + threadIdx.x;
  if (i >= ETOT * F) return;
  int eid = i / F, f = i - eid * F;
  int hh = f / D;
  int s, d;
  if (eid < NEDGES) { s = srcI[eid]; d = dstI[eid]; } else { s = eid - NEDGES; d = s; }
  atomicAdd(&out[(size_t)d * F + f], alpha[eid * H + hh] * h[(size_t)s * F + f]);
}

__global__ void bias_act(float* __restrict__ x, const float* __restrict__ bias,
                         int total, int F, int do_relu) {
  int i = blockIdx.x * blockDim.x + threadIdx.x;
  if (i >= total) return;
  float v = x[i] + bias[i % F];
  if (do_relu) v = fmaxf(v, 0.f);
  x[i] = v;
}

__global__ void pool_accum(const float* __restrict__ h, const int* __restrict__ batch,
                           float* __restrict__ sums, float* __restrict__ cnt) {
  int i = blockIdx.x * blockDim.x + threadIdx.x;
  if (i >= NNODES * F2) return;
  int n = i / F2, f = i - n * F2;
  int g = batch[n];
  atomicAdd(&sums[g * F2 + f], h[i]);
  if (f == 0) atomicAdd(&cnt[g], 1.0f);
}

__global__ void pool_final(const float* __restrict__ sums, const float* __restrict__ cnt,
                           float* __restrict__ out) {
  int i = blockIdx.x * blockDim.x + threadIdx.x;
  if (i >= NGRAPH * F2) return;
  out[i] = sums[i] / fmaxf(cnt[i / F2], 1.0f);
}

static inline int cdiv(long long a, int b) { return (int)((a + b - 1) / b); }

extern "C" void kernel_launch(void* const* d_in, const int* in_sizes, int n_in,
                              void* d_out, int out_size, void* d_ws, size_t ws_size,
                              hipStream_t stream) {
  const float* x        = (const float*)d_in[0];
  const int*   ei       = (const int*)d_in[1];     // [2, E] flat
  const int*   batch    = (const int*)d_in[2];
  const float* W1       = (const float*)d_in[3];
  const float* att_src1 = (const float*)d_in[4];
  const float* att_dst1 = (const float*)d_in[5];
  const float* bias1    = (const float*)d_in[6];
  const float* W2       = (const float*)d_in[7];
  const float* att_src2 = (const float*)d_in[8];
  const float* att_dst2 = (const float*)d_in[9];
  const float* bias2    = (const float*)d_in[10];
  float* out = (float*)d_out;
  const int* srcI = ei;
  const int* dstI = ei + NEDGES;

  // ---- workspace bump allocator (256B aligned) ----
  char* p = (char*)d_ws;
  auto alloc = [&](size_t bytes) -> char* {
    char* r = p; p += (bytes + 255) & ~(size_t)255; return r;
  };
  float*    h1    = (float*)alloc((size_t)NNODES * F1 * 4);
  float*    asrc1 = (float*)alloc((size_t)NNODES * H1C * 4);
  float*    adst1 = (float*)alloc((size_t)NNODES * H1C * 4);
  unsigned* mkey1 = (unsigned*)alloc((size_t)NNODES * H1C * 4);
  float*    ssum1 = (float*)alloc((size_t)NNODES * H1C * 4);
  float*    e1    = (float*)alloc((size_t)ETOT * H1C * 4);
  float*    agg1  = (float*)alloc((size_t)NNODES * F1 * 4);   // -> relu'd h
  float*    h2    = (float*)alloc((size_t)NNODES * F2 * 4);
  float*    asrc2 = (float*)alloc((size_t)NNODES * H2C * 4);
  float*    adst2 = (float*)alloc((size_t)NNODES * H2C * 4);
  unsigned* mkey2 = (unsigned*)alloc((size_t)NNODES * H2C * 4);
  float*    ssum2 = (float*)alloc((size_t)NNODES * H2C * 4);
  float*    e2    = (float*)alloc((size_t)ETOT * H2C * 4);
  float*    agg2  = (float*)alloc((size_t)NNODES * F2 * 4);
  float*    psum  = (float*)alloc((size_t)NGRAPH * F2 * 4);
  float*    pcnt  = (float*)alloc((size_t)NGRAPH * 4);

  // ---- clear accumulators (graph-capture-safe async memsets) ----
  hipMemsetAsync(mkey1, 0, (size_t)NNODES * H1C * 4, stream);
  hipMemsetAsync(ssum1, 0, (size_t)NNODES * H1C * 4, stream);
  hipMemsetAsync(agg1,  0, (size_t)NNODES * F1 * 4, stream);
  hipMemsetAsync(mkey2, 0, (size_t)NNODES * H2C * 4, stream);
  hipMemsetAsync(ssum2, 0, (size_t)NNODES * H2C * 4, stream);
  hipMemsetAsync(agg2,  0, (size_t)NNODES * F2 * 4, stream);
  hipMemsetAsync(psum,  0, (size_t)NGRAPH * F2 * 4, stream);
  hipMemsetAsync(pcnt,  0, (size_t)NGRAPH * 4, stream);

  const int T = 256;

  // ================= conv1 =================
  // h1 = x @ W1   [50000,128]x[128,384]
  wmma_gemm_f32<<<dim3(NNODES / 16, F1 / 64), 128, 0, stream>>>(x, W1, h1, FIN, F1);
  att_coeff<<<cdiv((long long)NNODES * H1C, T), T, 0, stream>>>(h1, att_src1, att_dst1,
                                                                asrc1, adst1, H1C, D1C);
  edge_logit_max<<<cdiv((long long)ETOT * H1C, T), T, 0, stream>>>(srcI, dstI, asrc1, adst1,
                                                                   e1, mkey1, H1C);
  edge_exp_sum<<<cdiv((long long)ETOT * H1C, T), T, 0, stream>>>(dstI, e1, mkey1, ssum1, H1C);
  edge_alpha<<<cdiv((long long)ETOT * H1C, T), T, 0, stream>>>(dstI, e1, ssum1, H1C);
  edge_agg<<<cdiv((long long)ETOT * F1, T), T, 0, stream>>>(srcI, dstI, e1, h1, agg1, H1C, D1C);
  bias_act<<<cdiv((long long)NNODES * F1, T), T, 0, stream>>>(agg1, bias1, NNODES * F1, F1, 1);

  // ================= conv2 =================
  // h2 = relu(conv1) @ W2   [50000,384]x[384,64]
  wmma_gemm_f32<<<dim3(NNODES / 16, F2 / 64), 128, 0, stream>>>(agg1, W2, h2, F1, F2);
  att_coeff<<<cdiv((long long)NNODES * H2C, T), T, 0, stream>>>(h2, att_src2, att_dst2,
                                                                asrc2, adst2, H2C, D2C);
  edge_logit_max<<<cdiv((long long)ETOT * H2C, T), T, 0, stream>>>(srcI, dstI, asrc2, adst2,
                                                                   e2, mkey2, H2C);
  edge_exp_sum<<<cdiv((long long)ETOT * H2C, T), T, 0, stream>>>(dstI, e2, mkey2, ssum2, H2C);
  edge_alpha<<<cdiv((long long)ETOT * H2C, T), T, 0, stream>>>(dstI, e2, ssum2, H2C);
  edge_agg<<<cdiv((long long)ETOT * F2, T), T, 0, stream>>>(srcI, dstI, e2, h2, agg2, H2C, D2C);
  bias_act<<<cdiv((long long)NNODES * F2, T), T, 0, stream>>>(agg2, bias2, NNODES * F2, F2, 0);

  // ================= mean pool =================
  pool_accum<<<cdiv((long long)NNODES * F2, T), T, 0, stream>>>(agg2, batch, psum, pcnt);
  pool_final<<<cdiv((long long)NGRAPH * F2, T), T, 0, stream>>>(psum, pcnt, out);
}